// BatchFrechetMean_16595753632498
// MI455X (gfx1250) — compile-verified
//
#include <hip/hip_runtime.h>

typedef float v2f __attribute__((ext_vector_type(2)));
typedef float v4f __attribute__((ext_vector_type(4)));
typedef float v8f __attribute__((ext_vector_type(8)));
typedef int   v4i __attribute__((ext_vector_type(4)));

// Explicit global-AS float so A/C traffic lowers to global_load/store (not flat)
typedef __attribute__((address_space(1))) float gfloat;
typedef __attribute__((address_space(1))) const v4f gc4f;

#define N        256
#define NN       (N * N)
#define NTHREADS 1024
#define NWAVES   32
#define NS_ITERS 9             // Newton-Schulz iterations per matrix sqrt
#define NBITS    6             // bits of t for the fractional power
#define NBUF     12

#define PANK     128           // K-panel: rows of B staged in LDS
#define PANELS   (N / PANK)    // 2
#define LDSW     (N + 8)       // padded LDS row stride (floats): 2*264 % 64 == 16
#define LDS_BYTES (PANK * LDSW * sizeof(float))   // 135168 B

#if defined(__gfx1250__) && __has_builtin(__builtin_amdgcn_global_load_async_to_lds_b128)
#define USE_ASYNC_LDS 1
typedef __attribute__((address_space(1))) v4i gv4i;   // global (prints as __device__)
typedef __attribute__((address_space(3))) v4i sv4i;   // LDS    (prints as __shared__)
#endif

// Fallback scratch if the harness workspace is too small (12 matrices = 3 MB)
__device__ float g_scratch[NBUF * NN];

__device__ __forceinline__ void wg_barrier() {
    __threadfence();
    __syncthreads();
}

// ---------------------------------------------------------------------------
// C = A * B (256x256x256 fp32) via V_WMMA_F32_16X16X4_F32.
// B is staged as a 128-row K-panel (full width) in LDS; accumulators persist
// across the two panels. Each wave owns 2 row-tiles x 4 col-tiles: per k-step
// 2 A b64 global loads + 8 LDS dwords feed 8 WMMAs (1.5 dwords/WMMA).
// ---------------------------------------------------------------------------
__device__ void wg_matmul(gfloat* __restrict__ C, const gfloat* __restrict__ A,
                          const gfloat* __restrict__ B, float* __restrict__ Bsh) {
    const int tid  = threadIdx.x;
    const int wave = tid >> 5;
    const int lane = tid & 31;
    const int l    = lane & 15;   // column / row within 16
    const int hi   = lane >> 4;   // lane half selects K (A/B) or M (C)
    const int rg   = wave >> 2;   // row group 0..7  -> rows rg*32 .. +31
    const int cg   = wave & 3;    // col group 0..3  -> cols cg*64 .. +63
    const int r0   = rg << 5;
    const int cc   = (cg << 6) + l;

    v8f acc00 = {0.f,0.f,0.f,0.f,0.f,0.f,0.f,0.f};
    v8f acc01 = acc00, acc02 = acc00, acc03 = acc00;
    v8f acc10 = acc00, acc11 = acc00, acc12 = acc00, acc13 = acc00;

    for (int p = 0; p < PANELS; ++p) {
        const int kb = p * PANK;

        __syncthreads();          // prior panel fully consumed before overwrite
        // ---- stage B K-panel (128 x 256 f32) into LDS: 8192 x b128 ----
        for (int fidx = tid; fidx < PANK * (N / 4); fidx += NTHREADS) {
            const int r = fidx >> 6;            // N/4 == 64 float4 per row
            const int q = (fidx & 63) << 2;
#if USE_ASYNC_LDS
            __builtin_amdgcn_global_load_async_to_lds_b128(
                (gv4i*)(B + (kb + r) * N + q),
                (sv4i*)(Bsh + r * LDSW + q),
                0, 0);
#else
            const v4f v = *(gc4f*)(B + (kb + r) * N + q);
            *(v4f*)(Bsh + r * LDSW + q) = v;
#endif
        }
#if USE_ASYNC_LDS
#if __has_builtin(__builtin_amdgcn_s_wait_asynccnt)
        __builtin_amdgcn_s_wait_asynccnt(0);
#else
        asm volatile("s_wait_asynccnt 0x0" ::: "memory");
#endif
#endif
        __syncthreads();

        // ---- accumulate 2 row-tiles x 4 col-tiles over this K-panel ----
        for (int kk = 0; kk < PANK; kk += 4) {
            const int ka = kk + (hi << 1);       // lane-half K within panel
            const int kg = kb + ka;              // global K for A columns
            v2f a0, a1, b0, b1, b2, b3;
            // A 16x4 fragments: lanes 0-15 K=kk,kk+1; lanes 16-31 K=kk+2,kk+3
            a0.x = A[(r0 + l) * N + kg];
            a0.y = A[(r0 + l) * N + kg + 1];
            a1.x = A[(r0 + 16 + l) * N + kg];
            a1.y = A[(r0 + 16 + l) * N + kg + 1];
            const float* brow0 = Bsh + ka * LDSW;
            const float* brow1 = brow0 + LDSW;
            b0.x = brow0[cc];      b0.y = brow1[cc];
            b1.x = brow0[cc + 16]; b1.y = brow1[cc + 16];
            b2.x = brow0[cc + 32]; b2.y = brow1[cc + 32];
            b3.x = brow0[cc + 48]; b3.y = brow1[cc + 48];
            acc00 = __builtin_amdgcn_wmma_f32_16x16x4_f32(false, a0, false, b0,
                                                          (short)0, acc00, false, false);
            acc01 = __builtin_amdgcn_wmma_f32_16x16x4_f32(false, a0, false, b1,
                                                          (short)0, acc01, false, false);
            acc02 = __builtin_amdgcn_wmma_f32_16x16x4_f32(false, a0, false, b2,
                                                          (short)0, acc02, false, false);
            acc03 = __builtin_amdgcn_wmma_f32_16x16x4_f32(false, a0, false, b3,
                                                          (short)0, acc03, false, false);
            acc10 = __builtin_amdgcn_wmma_f32_16x16x4_f32(false, a1, false, b0,
                                                          (short)0, acc10, false, false);
            acc11 = __builtin_amdgcn_wmma_f32_16x16x4_f32(false, a1, false, b1,
                                                          (short)0, acc11, false, false);
            acc12 = __builtin_amdgcn_wmma_f32_16x16x4_f32(false, a1, false, b2,
                                                          (short)0, acc12, false, false);
            acc13 = __builtin_amdgcn_wmma_f32_16x16x4_f32(false, a1, false, b3,
                                                          (short)0, acc13, false, false);
        }
    }

    // ---- store 8 C tiles ----
#pragma unroll
    for (int v = 0; v < 8; ++v) {
        gfloat* c0 = C + (r0 + (hi << 3) + v) * N + cc;
        c0[0]  = acc00[v];
        c0[16] = acc01[v];
        c0[32] = acc02[v];
        c0[48] = acc03[v];
        gfloat* c1 = C + (r0 + 16 + (hi << 3) + v) * N + cc;
        c1[0]  = acc10[v];
        c1[16] = acc11[v];
        c1[32] = acc12[v];
        c1[48] = acc13[v];
    }
    wg_barrier();
}

// ---------------------------------------------------------------------------
// Elementwise helpers (whole workgroup, uniform control flow, global AS)
// ---------------------------------------------------------------------------
__device__ void wg_axpbI(gfloat* __restrict__ D, const gfloat* __restrict__ S,
                         float alpha, float beta) {      // D = alpha*I + beta*S
    for (int i = threadIdx.x; i < NN; i += NTHREADS) {
        const int r = i >> 8, c = i & 255;
        float v = beta * S[i];
        if (r == c) v += alpha;
        D[i] = v;
    }
    wg_barrier();
}

__device__ void wg_scale(gfloat* __restrict__ D, const gfloat* __restrict__ S,
                         float s) {                      // D = s * S
    for (int i = threadIdx.x; i < NN; i += NTHREADS) D[i] = s * S[i];
    wg_barrier();
}

__device__ void wg_set_identity(gfloat* __restrict__ D) {
    for (int i = threadIdx.x; i < NN; i += NTHREADS) {
        const int r = i >> 8, c = i & 255;
        D[i] = (r == c) ? 1.0f : 0.0f;
    }
    wg_barrier();
}

__device__ void wg_sym(gfloat* __restrict__ D, const gfloat* __restrict__ S) {
    for (int i = threadIdx.x; i < NN; i += NTHREADS) {   // D = 0.5*(S + S^T)
        const int r = i >> 8, c = i & 255;
        D[i] = 0.5f * (S[i] + S[c * N + r]);
    }
    wg_barrier();
}

__device__ float wg_trace(const gfloat* __restrict__ S, float* red) {
    const int tid = threadIdx.x;
    if (tid < N) red[tid] = S[tid * N + tid];
    __syncthreads();
    for (int s = N / 2; s > 0; s >>= 1) {
        if (tid < s) red[tid] += red[tid + s];
        __syncthreads();
    }
    const float tr = red[0];
    __syncthreads();
    return tr;
}

// ---------------------------------------------------------------------------
// Coupled Newton-Schulz (Denman-Beavers):  SQ = Min^{1/2}, ISQ = Min^{-1/2}
// ---------------------------------------------------------------------------
__device__ void wg_sqrtm(const gfloat* __restrict__ Min, gfloat* SQ, gfloat* ISQ,
                         gfloat* Y, gfloat* Z, gfloat* T, gfloat* W,
                         float* red, float* Bsh) {
    float c = wg_trace(Min, red) * (1.0f / (float)N);    // mean eigenvalue
    c = fmaxf(c, 1e-6f);
    wg_scale(Y, Min, 1.0f / c);                          // Y0 = Min / c
    wg_set_identity(Z);                                  // Z0 = I
    for (int it = 0; it < NS_ITERS; ++it) {
        wg_matmul(W, Z, Y, Bsh);                         // W  = Z*Y
        wg_axpbI(T, W, 1.5f, -0.5f);                     // T  = 1.5 I - 0.5 W
        wg_matmul(W, Y, T, Bsh);                         // Y' = Y*T
        gfloat* tmp = Y; Y = W; W = tmp;
        wg_matmul(W, T, Z, Bsh);                         // Z' = T*Z
        tmp = Z; Z = W; W = tmp;
    }
    const float sc = sqrtf(c);
    wg_scale(SQ, Y, sc);                                 // sqrt(M)
    wg_scale(ISQ, Z, 1.0f / sc);                         // isqrt(M)
}

// ---------------------------------------------------------------------------
// Persistent single-workgroup scan over the batch (sequential dependency).
// ---------------------------------------------------------------------------
__global__ void __launch_bounds__(NTHREADS)
frechet_scan_kernel(const float* __restrict__ f,
                    const float* __restrict__ weights,
                    float* __restrict__ out,
                    float* __restrict__ ws, int B) {
    __shared__ float red[N];
    __shared__ float t_sh;
    extern __shared__ float Bsh[];     // 128 x 264 f32 K-panel buffer

    gfloat* wsg = (ws == nullptr) ? (gfloat*)&g_scratch[0] : (gfloat*)ws;
    const gfloat* fg = (const gfloat*)f;
    const gfloat* wg = (const gfloat*)weights;
    gfloat*       og = (gfloat*)out;

    gfloat* M   = wsg + 0  * NN;   // carry
    gfloat* SQ  = wsg + 1  * NN;
    gfloat* ISQ = wsg + 2  * NN;
    gfloat* S   = wsg + 3  * NN;
    gfloat* P   = wsg + 4  * NN;
    gfloat* R   = wsg + 5  * NN;
    gfloat* Y   = wsg + 6  * NN;   // NS scratch
    gfloat* Z   = wsg + 7  * NN;
    gfloat* T   = wsg + 8  * NN;
    gfloat* W   = wsg + 9  * NN;
    gfloat* T1  = wsg + 10 * NN;
    gfloat* T2  = wsg + 11 * NN;

    wg_set_identity(M);                                  // M0 = I

    for (int k = 0; k < B; ++k) {
        if (threadIdx.x == 0) {
            const float w0 = wg[2 * k], w1 = wg[2 * k + 1];
            t_sh = 1.0f / (1.0f + __expf(w0 - w1));      // softmax(w)[1]
        }
        __syncthreads();
        const float t = t_sh;
        __syncthreads();

        const gfloat* F = fg + (size_t)k * NN;

        // SQ = M^{1/2}, ISQ = M^{-1/2}
        wg_sqrtm(M, SQ, ISQ, Y, Z, T, W, red, Bsh);

        // S = sym(ISQ * F * ISQ)
        wg_matmul(T1, ISQ, F, Bsh);
        wg_matmul(T2, T1, ISQ, Bsh);
        wg_sym(S, T2);

        // P = S^t  via binary expansion: S^t = prod_j (S^{2^-j})^{b_j}
        wg_set_identity(P);
        wg_scale(R, S, 1.0f);
        float frac = t;
        for (int j = 0; j < NBITS; ++j) {
            wg_sqrtm(R, T1, T2, Y, Z, T, W, red, Bsh);   // T1 = sqrt(R)
            wg_scale(R, T1, 1.0f);
            frac *= 2.0f;
            if (frac >= 1.0f) {                          // uniform scalar branch
                frac -= 1.0f;
                wg_matmul(T1, P, R, Bsh);
                wg_scale(P, T1, 1.0f);
            }
        }

        // Mn = sym(SQ * P * SQ)
        wg_matmul(T1, SQ, P, Bsh);
        wg_matmul(T2, T1, SQ, Bsh);
        wg_sym(M, T2);

        // means[k] = Mn
        gfloat* dst = og + (size_t)k * NN;
        for (int i = threadIdx.x; i < NN; i += NTHREADS) dst[i] = M[i];
        wg_barrier();
    }
}

extern "C" void kernel_launch(void* const* d_in, const int* in_sizes, int n_in,
                              void* d_out, int out_size, void* d_ws, size_t ws_size,
                              hipStream_t stream) {
    const float* f = (const float*)d_in[0];   // (B,1,256,256) fp32
    const float* w = (const float*)d_in[1];   // (B,2) fp32
    float*       o = (float*)d_out;           // (B,1,256,256) fp32
    const int    B = in_sizes[1] / 2;

    float* ws = (ws_size >= (size_t)NBUF * NN * sizeof(float)) ? (float*)d_ws
                                                               : nullptr;
    frechet_scan_kernel<<<1, NTHREADS, LDS_BYTES, stream>>>(f, w, o, ws, B);
}